// Dropout_16587163697257
// MI455X (gfx1250) — compile-verified
//
#include <hip/hip_runtime.h>
#include <stdint.h>

// Dropout-by-index: out = X; out[drop_idx] = 0
// Phase 1: streaming copy X -> out via CDNA5 async global<->LDS DMA (b128),
//          X read with TH_LOAD_NT so the 134MB one-shot stream does not evict
//          `out` (134MB, L2-resident: global L2 = 192MB).
// Phase 2: request-bound scatter of 0.0f into out (hits L2); index stream
//          read non-temporally (one-shot 80.5MB).

#define CPB   256   // threads per block = 8 wave32
#define SLOTS 8     // float4 tiles per thread -> 128B/thread, 32KB/block

#if defined(__gfx1250__)
__device__ __forceinline__ void async_wait0() {
#if __has_builtin(__builtin_amdgcn_s_wait_asynccnt)
  __builtin_amdgcn_s_wait_asynccnt(0);
#else
  asm volatile("s_wait_asynccnt 0" ::: "memory");
#endif
}

// lds_off: wave-relative LDS byte address (low 32 bits of flat pointer to __shared__)
__device__ __forceinline__ void async_load_b128_nt(uint32_t lds_off, const void* gaddr) {
  asm volatile("global_load_async_to_lds_b128 %0, %1, off th:TH_LOAD_NT"
               :
               : "v"(lds_off), "v"((uint64_t)(uintptr_t)gaddr)
               : "memory");
}

__device__ __forceinline__ void async_store_b128(void* gaddr, uint32_t lds_off) {
  asm volatile("global_store_async_from_lds_b128 %0, %1, off"
               :
               : "v"((uint64_t)(uintptr_t)gaddr), "v"(lds_off)
               : "memory");
}
#endif  // __gfx1250__

__global__ __launch_bounds__(CPB)
void dropout_copy_kernel(const float4* __restrict__ src,
                         float4* __restrict__ dst,
                         long long n4) {
  const long long base = (long long)blockIdx.x * (CPB * SLOTS) + threadIdx.x;

#if defined(__gfx1250__)
  __shared__ float4 stage[CPB * SLOTS];
  // Full-tile fast path: all slots in range -> EXEC all-ones for async ops.
  if (base + (long long)(SLOTS - 1) * CPB < n4) {
    const uint32_t l0 = (uint32_t)(uintptr_t)(&stage[threadIdx.x]);
#pragma unroll
    for (int s = 0; s < SLOTS; ++s) {
      async_load_b128_nt(l0 + (uint32_t)(s * CPB * sizeof(float4)),
                         (const void*)(src + base + (long long)s * CPB));
    }
    async_wait0();  // loads into LDS complete
#pragma unroll
    for (int s = 0; s < SLOTS; ++s) {
      async_store_b128((void*)(dst + base + (long long)s * CPB),
                       l0 + (uint32_t)(s * CPB * sizeof(float4)));
    }
    async_wait0();  // stores drained before wave retires (ENDPGM also waits)
    return;
  }
#endif
  // Tail / non-gfx1250 fallback: plain 128-bit copies.
#pragma unroll
  for (int s = 0; s < SLOTS; ++s) {
    const long long i = base + (long long)s * CPB;
    if (i < n4) dst[i] = src[i];
  }
}

__global__ __launch_bounds__(256)
void dropout_scatter_kernel(const long long* __restrict__ drop_idx,
                            float* __restrict__ out,
                            long long k) {
  long long i = (long long)blockIdx.x * blockDim.x + threadIdx.x;
  const long long stride = (long long)gridDim.x * blockDim.x;
  for (; i < k; i += stride) {
    const long long j = __builtin_nontemporal_load(&drop_idx[i]);  // one-shot stream
    out[j] = 0.0f;  // random 4B store; out is L2-resident after phase 1
  }
}

extern "C" void kernel_launch(void* const* d_in, const int* in_sizes, int n_in,
                              void* d_out, int out_size, void* d_ws, size_t ws_size,
                              hipStream_t stream) {
  const float*     X    = (const float*)d_in[0];
  const long long* didx = (const long long*)d_in[1];
  float*           out  = (float*)d_out;

  const long long n = (long long)in_sizes[0];  // 8192*4096 = 2^25
  const long long k = (long long)in_sizes[1];  // ~10.07M

  // Phase 1: copy. n4 = 2^23 is exactly divisible by the 2048-float4 tile,
  // so every block takes the uniform async fast path (no tail in practice;
  // the in-kernel guarded scalar path covers any other shape).
  const long long n4 = n / 4;
  const long long tile = (long long)CPB * SLOTS;  // 2048 float4 per block
  const int copy_blocks = (int)((n4 + tile - 1) / tile);
  dropout_copy_kernel<<<copy_blocks, CPB, 0, stream>>>(
      (const float4*)X, (float4*)out, n4);

  // Phase 2: scatter zeros (ordered after phase 1 by stream semantics).
  const int sc_blocks = (int)((k + 255) / 256);
  dropout_scatter_kernel<<<sc_blocks, 256, 0, stream>>>(didx, out, k);
}